// L_ZL_R_87540023427866
// MI455X (gfx1250) — compile-verified
//
#include <hip/hip_runtime.h>
#include <hip/hip_bf16.h>

// ---------------------------------------------------------------------------
// Problem constants (fixed by the reference's setup_inputs):
//   b=16, w=480, h=640 ; M=8 spatial blocks/side ; D=10 depth-bin width
//   NK = 26 depth bins ; nseg = 8*8*26 = 1664 ; NUM_CLASS = 40
// ---------------------------------------------------------------------------
#define IGNORE_LABEL 255
#define NUM_CLASS    40
#define MBLK         8
#define NKBIN        26
#define NSEG         (MBLK * MBLK * NKBIN)   // 1664
#define WDIM         480
#define HDIM         640
#define BW           (WDIM / MBLK)           // 60
#define BH           (HDIM / MBLK)           // 80
#define TPB          256

// Workspace layout (dwords):
//   [0..3]                     scalar accumulators: sum_ld, sum_ld2, sum_lr, n_valid
//   [4            .. 4+  NSEG) per-seg pixel count   (f32)
//   [4+  NSEG     .. 4+2*NSEG) per-seg loss_1 sum    (f32)
//   [4+2*NSEG     .. 4+4*NSEG) per-seg gt-class presence bits   (2x u32)
//   [4+4*NSEG     .. 4+6*NSEG) per-seg pred-class presence bits (2x u32)
#define WS_DWORDS (4 + 6 * NSEG)             // 9988 dwords (~39 KB)

__global__ __launch_bounds__(TPB) void zero_ws_kernel(float* __restrict__ ws) {
    int i = blockIdx.x * blockDim.x + threadIdx.x;
    if (i < WS_DWORDS) ws[i] = 0.0f;         // bitwise zero also clears u32 regions
}

__device__ __forceinline__ int clampi(int v, int lo, int hi) {
    return v < lo ? lo : (v > hi ? hi : v);
}

// Truncate a generic pointer to an LDS-space object down to its 32-bit LDS
// byte address (generic LDS addresses carry the DS offset in addr[31:0]).
__device__ __forceinline__ unsigned lds_addr32(const void* p) {
    return (unsigned)(unsigned long long)p;
}

// CDNA5 async copy: 16 bytes from global base + voff to LDS byte address lds,
// tracked by ASYNCcnt (fire-and-forget; no VGPR destination).
__device__ __forceinline__ void async_copy_b128(unsigned lds, const void* base, unsigned voff) {
    asm volatile("global_load_async_to_lds_b128 %0, %1, %2"
                 :: "v"(lds), "v"(voff), "s"(base)
                 : "memory");
}
__device__ __forceinline__ void wait_async_le5() {
    asm volatile("s_wait_asynccnt 0x5" ::: "memory");
}
__device__ __forceinline__ void wait_async_0() {
    asm volatile("s_wait_asynccnt 0x0" ::: "memory");
}

__global__ __launch_bounds__(TPB) void main_pass_kernel(
    const int*   __restrict__ pred,
    const int*   __restrict__ heat,
    const float* __restrict__ depth,
    const float* __restrict__ loss1,
    const float* __restrict__ pdepth,
    float*       __restrict__ ws,
    int n4)                                   // N / 4 float4 groups
{
    // ---- LDS-privatized segment statistics (one copy per workgroup) ----
    __shared__ float    sCount[NSEG];
    __shared__ float    sLoss [NSEG];
    __shared__ unsigned sPL   [2 * NSEG];
    __shared__ unsigned sPP   [2 * NSEG];
    __shared__ float    sScal [4];

    // ---- double-buffered async staging: 16 B per thread per stream ----
    __shared__ alignas(16) float bufD[2][TPB * 4];
    __shared__ alignas(16) float bufP[2][TPB * 4];
    __shared__ alignas(16) float bufL[2][TPB * 4];
    __shared__ alignas(16) int   bufH[2][TPB * 4];
    __shared__ alignas(16) int   bufR[2][TPB * 4];

    for (int s = threadIdx.x; s < NSEG; s += blockDim.x) { sCount[s] = 0.0f; sLoss[s] = 0.0f; }
    for (int s = threadIdx.x; s < 2 * NSEG; s += blockDim.x) { sPL[s] = 0u; sPP[s] = 0u; }
    if (threadIdx.x < 4) sScal[threadIdx.x] = 0.0f;
    __syncthreads();

    const unsigned t16 = threadIdx.x * 16u;
    const unsigned aD[2] = { lds_addr32(&bufD[0][0]) + t16, lds_addr32(&bufD[1][0]) + t16 };
    const unsigned aP[2] = { lds_addr32(&bufP[0][0]) + t16, lds_addr32(&bufP[1][0]) + t16 };
    const unsigned aL[2] = { lds_addr32(&bufL[0][0]) + t16, lds_addr32(&bufL[1][0]) + t16 };
    const unsigned aH[2] = { lds_addr32(&bufH[0][0]) + t16, lds_addr32(&bufH[1][0]) + t16 };
    const unsigned aR[2] = { lds_addr32(&bufR[0][0]) + t16, lds_addr32(&bufR[1][0]) + t16 };

    float acc_ld = 0.0f, acc_ld2 = 0.0f, acc_lr = 0.0f, acc_valid = 0.0f;

    const int tid    = blockIdx.x * blockDim.x + threadIdx.x;
    const int stride = gridDim.x * blockDim.x;

    // Prologue: stage this thread's first tile into buffer 0.
    if (tid < n4) {
        const unsigned voff = (unsigned)tid * 16u;
        async_copy_b128(aD[0], depth,  voff);
        async_copy_b128(aP[0], pdepth, voff);
        async_copy_b128(aL[0], loss1,  voff);
        async_copy_b128(aH[0], heat,   voff);
        async_copy_b128(aR[0], pred,   voff);
    }

    int k = 0;
    for (int i = tid; i < n4; i += stride, k ^= 1) {
        const int inext = i + stride;
        if (inext < n4) {
            // Issue the next tile's copies, then wait only for the current
            // tile (async loads complete in order -> asynccnt <= 5).
            const unsigned voff = (unsigned)inext * 16u;
            async_copy_b128(aD[k ^ 1], depth,  voff);
            async_copy_b128(aP[k ^ 1], pdepth, voff);
            async_copy_b128(aL[k ^ 1], loss1,  voff);
            async_copy_b128(aH[k ^ 1], heat,   voff);
            async_copy_b128(aR[k ^ 1], pred,   voff);
            wait_async_le5();
        } else {
            wait_async_0();
        }

        const float4 d4 = *reinterpret_cast<const float4*>(&bufD[k][threadIdx.x * 4]);
        const float4 p4 = *reinterpret_cast<const float4*>(&bufP[k][threadIdx.x * 4]);
        const float4 l4 = *reinterpret_cast<const float4*>(&bufL[k][threadIdx.x * 4]);
        const int4   h4 = *reinterpret_cast<const int4*  >(&bufH[k][threadIdx.x * 4]);
        const int4   r4 = *reinterpret_cast<const int4*  >(&bufR[k][threadIdx.x * 4]);

        const float dd[4] = {d4.x, d4.y, d4.z, d4.w};
        const float pp[4] = {p4.x, p4.y, p4.z, p4.w};
        const float ll[4] = {l4.x, l4.y, l4.z, l4.w};
        const int   hh[4] = {h4.x, h4.y, h4.z, h4.w};
        const int   rr[4] = {r4.x, r4.y, r4.z, r4.w};

        // 4 consecutive pixels: base%4==0, HDIM%4==0 and BH%4==0, so this
        // group never crosses a row boundary or an 80-col sub-block boundary.
        const int base = i * 4;
        const int rem  = base % (WDIM * HDIM);
        const int row  = rem / HDIM;
        const int col0 = rem - row * HDIM;
        const bool row_ok  = (row % BW) < (BW - 1);
        const int  cmod    = col0 % BH;                       // in {0,4,...,76}
        const int  segBase = ((row / BW) * MBLK + (col0 / BH)) * NKBIN;

#pragma unroll
        for (int j = 0; j < 4; ++j) {
            const float d = dd[j];
            const float L = ll[j];
            const int   hv = hh[j];
            const int   pv = rr[j];

            const float ld = logf(fmaf(pp[j], 255.0f, 1.0f)) - logf(d + 1.0f);
            acc_ld  += ld;
            acc_ld2  = fmaf(ld, ld, acc_ld2);
            acc_lr   = fmaf(L, fabsf(ld), acc_lr);

            const bool valid = (hv != IGNORE_LABEL);
            acc_valid += valid ? 1.0f : 0.0f;

            // depth bin: k*10 < d < k*10+9 (strict), matching the reference
            const float kfF = floorf(d / 10.0f);
            const int   kf  = (int)kfF;
            const float k10 = kfF * 10.0f;
            const bool k_ok = (d > k10) && (d < k10 + 9.0f) && (kf >= 0) && (kf < NKBIN);
            const bool ok   = valid && row_ok && ((cmod + j) < (BH - 1)) && k_ok;

            if (ok) {
                const int seg = segBase + kf;                 // kf in [0,25]
                atomicAdd(&sCount[seg], 1.0f);                // ds_add_f32
                atomicAdd(&sLoss[seg], L);
                const int lh = clampi(hv, 0, NUM_CLASS - 1);
                const int lp = clampi(pv, 0, NUM_CLASS - 1);
                atomicOr(&sPL[2 * seg + (lh >> 5)], 1u << (lh & 31));
                atomicOr(&sPP[2 * seg + (lp >> 5)], 1u << (lp & 31));
            }
        }
    }

    // ---- wave32 shuffle reduction of the four scalar accumulators ----
#pragma unroll
    for (int off = 16; off > 0; off >>= 1) {
        acc_ld    += __shfl_xor(acc_ld,    off, 32);
        acc_ld2   += __shfl_xor(acc_ld2,   off, 32);
        acc_lr    += __shfl_xor(acc_lr,    off, 32);
        acc_valid += __shfl_xor(acc_valid, off, 32);
    }
    if ((threadIdx.x & 31) == 0) {
        atomicAdd(&sScal[0], acc_ld);
        atomicAdd(&sScal[1], acc_ld2);
        atomicAdd(&sScal[2], acc_lr);
        atomicAdd(&sScal[3], acc_valid);
    }
    __syncthreads();
    if (threadIdx.x < 4) atomicAdd(&ws[threadIdx.x], sScal[threadIdx.x]);

    // ---- flush LDS-private histograms to global accumulators ----
    float*    gCount = ws + 4;
    float*    gLoss  = ws + 4 + NSEG;
    unsigned* gPL    = (unsigned*)(ws + 4 + 2 * NSEG);
    unsigned* gPP    = (unsigned*)(ws + 4 + 4 * NSEG);
    for (int s = threadIdx.x; s < NSEG; s += blockDim.x) {
        const float c = sCount[s];
        if (c != 0.0f) {
            atomicAdd(&gCount[s], c);
            atomicAdd(&gLoss[s], sLoss[s]);
        }
        const unsigned a0 = sPL[2 * s], a1 = sPL[2 * s + 1];
        const unsigned b0 = sPP[2 * s], b1 = sPP[2 * s + 1];
        if (a0) atomicOr(&gPL[2 * s],     a0);
        if (a1) atomicOr(&gPL[2 * s + 1], a1);
        if (b0) atomicOr(&gPP[2 * s],     b0);
        if (b1) atomicOr(&gPP[2 * s + 1], b1);
    }
}

__global__ __launch_bounds__(TPB) void finalize_kernel(
    const float* __restrict__ ws, float* __restrict__ out, float invN)
{
    const float*    gCount = ws + 4;
    const float*    gLoss  = ws + 4 + NSEG;
    const unsigned* gPL    = (const unsigned*)(ws + 4 + 2 * NSEG);
    const unsigned* gPP    = (const unsigned*)(ws + 4 + 4 * NSEG);

    float loss = 0.0f, times = 0.0f;
    for (int s = threadIdx.x; s < NSEG; s += blockDim.x) {
        const float c = gCount[s];
        if (c > 0.0f) {
            // |pres_l - pres_p| summed over 40 classes == popcount(XOR)
            const float diff = (float)(__popc(gPL[2 * s]     ^ gPP[2 * s]) +
                                       __popc(gPL[2 * s + 1] ^ gPP[2 * s + 1]));
            loss  += diff * (gLoss[s] / fmaxf(c, 1.0f));
            times += 1.0f;
        }
    }
#pragma unroll
    for (int off = 16; off > 0; off >>= 1) {
        loss  += __shfl_xor(loss,  off, 32);
        times += __shfl_xor(times, off, 32);
    }
    __shared__ float sh[2];
    if (threadIdx.x == 0) { sh[0] = 0.0f; sh[1] = 0.0f; }
    __syncthreads();
    if ((threadIdx.x & 31) == 0) { atomicAdd(&sh[0], loss); atomicAdd(&sh[1], times); }
    __syncthreads();

    if (threadIdx.x == 0) {
        const float sld  = ws[0];
        const float sld2 = ws[1];
        const float slr  = ws[2];
        const float sval = ws[3];
        const float m = sld * invN;
        out[0] = sld2 * invN - 0.5f * m * m;   // data_loss = s1 - 0.5*s2
        out[1] = slr / sval;                   // LR
        out[2] = sh[0] / (sh[1] + 0.001f);     // loss / (times + 0.001)
    }
}

extern "C" void kernel_launch(void* const* d_in, const int* in_sizes, int n_in,
                              void* d_out, int out_size, void* d_ws, size_t ws_size,
                              hipStream_t stream) {
    const int*   pred   = (const int*)  d_in[0];
    const int*   heat   = (const int*)  d_in[1];
    const float* depth  = (const float*)d_in[2];
    const float* loss1  = (const float*)d_in[3];
    const float* pdepth = (const float*)d_in[4];
    float* ws  = (float*)d_ws;
    float* out = (float*)d_out;

    const int N  = in_sizes[0];               // b*w*h = 4,915,200 (divisible by 4)
    const int n4 = N / 4;

    zero_ws_kernel<<<(WS_DWORDS + 255) / 256, 256, 0, stream>>>(ws);

    // 256 blocks x 256 threads = 2048 wave32s; ~19 float4 iterations/thread.
    main_pass_kernel<<<256, TPB, 0, stream>>>(pred, heat, depth, loss1, pdepth, ws, n4);

    finalize_kernel<<<1, TPB, 0, stream>>>(ws, out, 1.0f / (float)N);
}